// GATModel_89902255439933
// MI455X (gfx1250) — compile-verified
//
#include <hip/hip_runtime.h>
#include <math.h>
#include <stdint.h>

// ---------------------------------------------------------------------------
// 2-layer GAT for MI455X (gfx1250, wave32).
//  - GEMMs: V_WMMA_F32_16X16X4_F32. B panel staged in LDS per workgroup
//    (async global->LDS DMA when available, ASYNCcnt + barrier), bank-conflict
//    -free padded LDS layout, A streamed as v2f (global_load_b64), K-loop
//    unrolled (K is a template constant).
//  - Edge softmax/aggregation: 3-pass segment softmax with float atomics
//    (ordered-int atomicMax for segment_max), biases folded into accumulator
//    init, ELU in place.  Hot accumulators (~65 MB) stay in the 192 MB L2.
// ---------------------------------------------------------------------------

typedef __attribute__((ext_vector_type(2))) float v2f;
typedef __attribute__((ext_vector_type(8))) float v8f;
typedef __attribute__((ext_vector_type(4))) int v4i;

#define NEG_SLOPE 0.2f

#if defined(__gfx1250__) && __has_builtin(__builtin_amdgcn_global_load_async_to_lds_b128)
#define USE_ASYNC_LDS 1
#else
#define USE_ASYNC_LDS 0
#endif

__device__ __forceinline__ int f2ord(float f) {
  int i = __float_as_int(f);
  return i >= 0 ? i : (i ^ 0x7fffffff);
}
__device__ __forceinline__ float ord2f(int i) {
  return __int_as_float(i >= 0 ? i : (i ^ 0x7fffffff));
}

// ---------------- init kernels ----------------
__global__ void fill_f32_kernel(float* __restrict__ p, float v, long long n) {
  long long i = (long long)blockIdx.x * blockDim.x + threadIdx.x;
  if (i < n) p[i] = v;
}

__global__ void fill_i32_kernel(int* __restrict__ p, int v, long long n) {
  long long i = (long long)blockIdx.x * blockDim.x + threadIdx.x;
  if (i < n) p[i] = v;
}

// out[n*C + c] = bias[c]  (folds "+ bias" into the scatter accumulator)
__global__ void init_bias_kernel(float* __restrict__ p, const float* __restrict__ b,
                                 int C, long long n) {
  long long i = (long long)blockIdx.x * blockDim.x + threadIdx.x;
  if (i < n) p[i] = b[(int)(i % C)];
}

// pack B[K,Nreal] into Bpad[K,16] with zero fill (for layer-2 N=10)
__global__ void pad_b_kernel(const float* __restrict__ B, float* __restrict__ Bp,
                             int K, int Nreal, int Npad) {
  int i = blockIdx.x * blockDim.x + threadIdx.x;  // over K*Npad
  if (i >= K * Npad) return;
  int k = i / Npad, c = i - k * Npad;
  Bp[i] = (c < Nreal) ? B[k * Nreal + c] : 0.f;
}

// ---------------- WMMA f32 GEMM with LDS-staged B --------------------------
// Grid: blockIdx.x = group of 8 row-tiles (one 16-row tile per wave),
//       blockIdx.y = N-strip of width W = 16*NB shared by the whole block.
// B[:, strip] is staged into LDS once (async DMA if available), then each
// wave runs the K loop reading B fragments from LDS and streaming A (v2f).
// LDS row stride WP = W+24 -> the two half-wave reads (rows k+0/k+2, cols
// 0-15 vs 16-31) hit disjoint bank groups: conflict-free.
template <int NB, int K>
__global__ __launch_bounds__(256) void gemm_wmma_lds(
    const float* __restrict__ A, const float* __restrict__ B,
    float* __restrict__ C, int M, int NpadB, int Nreal, int ldc) {
  constexpr int W = 16 * NB;
  constexpr int WP = W + 24;  // padded LDS row stride (floats)
  __shared__ float Bs[K * WP];

  const int lane = threadIdx.x & 31;
  const int wave = threadIdx.x >> 5;
  const int tn = blockIdx.y;

  // ---- stage B[:, tn*W .. tn*W+W) into LDS, 16B granules ----
  {
    const float* Bg = B + (size_t)tn * W;
    constexpr int TOT4 = K * W / 4;  // float4 granules
    for (int idx = threadIdx.x; idx < TOT4; idx += 256) {
      int r = idx / (W / 4);
      int c4 = idx - r * (W / 4);
      const float* src = Bg + (size_t)r * NpadB + c4 * 4;
      float* dst = Bs + r * WP + c4 * 4;  // 16B aligned: WP%4==0
#if USE_ASYNC_LDS
      __builtin_amdgcn_global_load_async_to_lds_b128(
          (__attribute__((address_space(1))) v4i*)(src),
          (__attribute__((address_space(3))) v4i*)(dst), 0, 0);
#else
      *(float4*)dst = *(const float4*)src;
#endif
    }
#if USE_ASYNC_LDS
#if __has_builtin(__builtin_amdgcn_s_wait_asynccnt)
    __builtin_amdgcn_s_wait_asynccnt(0);
#else
    asm volatile("s_wait_asynccnt 0x0" ::: "memory");
#endif
#endif
    __syncthreads();
  }

  const int tilesM = (M + 15) >> 4;
  const int tm = blockIdx.x * 8 + wave;
  if (tm >= tilesM) return;  // uniform per wave; after barrier

  int row = tm * 16 + (lane & 15);
  int rowL = row < M ? row : (M - 1);  // clamp: no EXEC divergence
  const int kbase = (lane < 16) ? 0 : 2;

  v8f acc[NB];
#pragma unroll
  for (int j = 0; j < NB; ++j) acc[j] = (v8f){0.f,0.f,0.f,0.f,0.f,0.f,0.f,0.f};

  const float* Ap = A + (size_t)rowL * K + kbase;  // 8B aligned (kbase even)

#pragma unroll 4
  for (int k = 0; k < K; k += 4) {
    v2f a = *(const v2f*)Ap;
    Ap += 4;
    const float* bl = Bs + (k + kbase) * WP + (lane & 15);
#pragma unroll
    for (int j = 0; j < NB; ++j) {
      v2f b;
      b.x = bl[j * 16];       // row k+kbase
      b.y = bl[WP + j * 16];  // row k+kbase+1
      acc[j] = __builtin_amdgcn_wmma_f32_16x16x4_f32(false, a, false, b,
                                                     (short)0, acc[j], false,
                                                     false);
    }
  }

  int mbase = tm * 16 + ((lane < 16) ? 0 : 8);
#pragma unroll
  for (int j = 0; j < NB; ++j) {
    int col = tn * W + j * 16 + (lane & 15);
    if (col < Nreal) {
#pragma unroll
      for (int v = 0; v < 8; ++v) {
        int r = mbase + v;
        if (r < M) C[(size_t)r * ldc + col] = acc[j][v];
      }
    }
  }
}

// ---------------- attention dot products: a_src/a_dst [N,H] ----------------
__global__ void att_dots_kernel(const float* __restrict__ h,
                                const float* __restrict__ att_src,
                                const float* __restrict__ att_dst,
                                float* __restrict__ aS, float* __restrict__ aD,
                                int N, int H, int Cc) {
  int i = blockIdx.x * blockDim.x + threadIdx.x;  // over N*H
  if (i >= N * H) return;
  int n = i / H, hh = i - n * H;
  const float* hp = h + (size_t)n * H * Cc + (size_t)hh * Cc;
  const float* as = att_src + hh * Cc;
  const float* ad = att_dst + hh * Cc;
  float s1 = 0.f, s2 = 0.f;
  for (int c = 0; c < Cc; ++c) {
    float v = hp[c];
    s1 += v * as[c];
    s2 += v * ad[c];
  }
  aS[i] = s1;
  aD[i] = s2;
}

// edge e < E: (src,dst) from edge_index; e >= E: self loop (e-E, e-E)
__device__ __forceinline__ void edge_nodes(const int* __restrict__ ei, int E,
                                           int e, int& s, int& d) {
  if (e < E) { s = ei[e]; d = ei[E + e]; }
  else       { s = e - E; d = e - E; }
}

// pass 1: logits + segment max (ordered-int atomicMax)
__global__ void edge_logit_max_kernel(const int* __restrict__ ei, int E, int N,
                                      const float* __restrict__ aS,
                                      const float* __restrict__ aD,
                                      int* __restrict__ m, int H) {
  long long i = (long long)blockIdx.x * blockDim.x + threadIdx.x;
  long long total = (long long)(E + N) * H;
  if (i >= total) return;
  int e = (int)(i / H), hh = (int)(i - (long long)e * H);
  int s, d;
  edge_nodes(ei, E, e, s, d);
  float l = aS[(size_t)s * H + hh] + aD[(size_t)d * H + hh];
  l = l > 0.f ? l : l * NEG_SLOPE;
  atomicMax(&m[(size_t)d * H + hh], f2ord(l));
}

// pass 2: e = exp(logit - max); store e per edge; segment sum via atomicAdd
__global__ void edge_exp_sum_kernel(const int* __restrict__ ei, int E, int N,
                                    const float* __restrict__ aS,
                                    const float* __restrict__ aD,
                                    const int* __restrict__ m,
                                    float* __restrict__ ebuf,
                                    float* __restrict__ ssum, int H) {
  long long i = (long long)blockIdx.x * blockDim.x + threadIdx.x;
  long long total = (long long)(E + N) * H;
  if (i >= total) return;
  int e = (int)(i / H), hh = (int)(i - (long long)e * H);
  int s, d;
  edge_nodes(ei, E, e, s, d);
  float l = aS[(size_t)s * H + hh] + aD[(size_t)d * H + hh];
  l = l > 0.f ? l : l * NEG_SLOPE;
  float mf = ord2f(m[(size_t)d * H + hh]);
  float ex = __expf(l - mf);
  ebuf[i] = ex;
  atomicAdd(&ssum[(size_t)d * H + hh], ex);
}

// pass 3: out[dst, c] += (e/s) * h[src, c]   (thread per (edge, channel))
__global__ void edge_aggregate_kernel(const int* __restrict__ ei, int E, int N,
                                      const float* __restrict__ ebuf,
                                      const float* __restrict__ ssum,
                                      const float* __restrict__ h,
                                      float* __restrict__ out, int H, int Cc) {
  long long i = (long long)blockIdx.x * blockDim.x + threadIdx.x;
  int HC = H * Cc;
  long long total = (long long)(E + N) * HC;
  if (i >= total) return;
  int e = (int)(i / HC);
  int c = (int)(i - (long long)e * HC);
  int hh = c / Cc;
  int s, d;
  edge_nodes(ei, E, e, s, d);
  float alpha = ebuf[(size_t)e * H + hh] / ssum[(size_t)d * H + hh];
  atomicAdd(&out[(size_t)d * HC + c], alpha * h[(size_t)s * HC + c]);
}

// ELU in place (bias was already folded into the accumulator init)
__global__ void elu_inplace_kernel(float* __restrict__ p, long long n) {
  long long i = (long long)blockIdx.x * blockDim.x + threadIdx.x;
  if (i < n) {
    float v = p[i];
    p[i] = v > 0.f ? v : (__expf(v) - 1.f);
  }
}

// ---------------------------------------------------------------------------
static inline unsigned nblk(long long n) { return (unsigned)((n + 255) / 256); }

extern "C" void kernel_launch(void* const* d_in, const int* in_sizes, int n_in,
                              void* d_out, int out_size, void* d_ws, size_t ws_size,
                              hipStream_t stream) {
  const float* x   = (const float*)d_in[0];
  const int*   ei  = (const int*)d_in[1];   // [2,E] int32 (JAX x64 disabled)
  const float* W1  = (const float*)d_in[2]; // [128, 256]
  const float* as1 = (const float*)d_in[3]; // [4, 64]
  const float* ad1 = (const float*)d_in[4];
  const float* b1  = (const float*)d_in[5]; // [256]
  const float* W2  = (const float*)d_in[6]; // [256, 10]
  const float* as2 = (const float*)d_in[7]; // [1, 10]
  const float* ad2 = (const float*)d_in[8];
  const float* b2  = (const float*)d_in[9]; // [10]

  const int F = 128, H = 4, C1 = 64, HC1 = 256, NC = 10;
  const int N = in_sizes[0] / F;
  const int E = in_sizes[1] / 2;
  const int Etot = E + N;

  // workspace layout
  char* w = (char*)d_ws;
  float* h1   = (float*)w; w += (size_t)N * HC1 * 4;   // x @ W1
  float* out1 = (float*)w; w += (size_t)N * HC1 * 4;   // layer-1 agg -> elu
  float* aS1  = (float*)w; w += (size_t)N * H * 4;
  float* aD1  = (float*)w; w += (size_t)N * H * 4;
  int*   m1   = (int*)w;   w += (size_t)N * H * 4;
  float* s1   = (float*)w; w += (size_t)N * H * 4;
  float* e1   = (float*)w; w += (size_t)Etot * H * 4;
  float* h2   = (float*)w; w += (size_t)N * NC * 4;    // elu(out1) @ W2
  float* aS2  = (float*)w; w += (size_t)N * 4;
  float* aD2  = (float*)w; w += (size_t)N * 4;
  int*   m2   = (int*)w;   w += (size_t)N * 4;
  float* s2   = (float*)w; w += (size_t)N * 4;
  float* e2   = (float*)w; w += (size_t)Etot * 4;
  float* W2p  = (float*)w; w += (size_t)HC1 * 16 * 4;  // W2 padded to [256,16]
  float* out2 = (float*)d_out;                         // [N, 10]

  // f2ord(-inf): bits(-inf)=0xFF800000 (<0) -> ^0x7FFFFFFF = 0x807FFFFF
  const int mInit = (int)0x807fffff;
  const int tilesM = (N + 15) / 16;

  // ---------------- layer 1 ----------------
  {
    dim3 grid((tilesM + 7) / 8, HC1 / 64);  // NB=4 -> W=64 per strip
    gemm_wmma_lds<4, 128><<<grid, 256, 0, stream>>>(x, W1, h1, N, HC1, HC1,
                                                    HC1);
  }
  att_dots_kernel<<<nblk((long long)N * H), 256, 0, stream>>>(h1, as1, ad1, aS1,
                                                              aD1, N, H, C1);
  fill_i32_kernel<<<nblk((long long)N * H), 256, 0, stream>>>(m1, mInit,
                                                              (long long)N * H);
  fill_f32_kernel<<<nblk((long long)N * H), 256, 0, stream>>>(s1, 0.f,
                                                              (long long)N * H);
  init_bias_kernel<<<nblk((long long)N * HC1), 256, 0, stream>>>(
      out1, b1, HC1, (long long)N * HC1);

  edge_logit_max_kernel<<<nblk((long long)Etot * H), 256, 0, stream>>>(
      ei, E, N, aS1, aD1, m1, H);
  edge_exp_sum_kernel<<<nblk((long long)Etot * H), 256, 0, stream>>>(
      ei, E, N, aS1, aD1, m1, e1, s1, H);
  edge_aggregate_kernel<<<nblk((long long)Etot * HC1), 256, 0, stream>>>(
      ei, E, N, e1, s1, h1, out1, H, C1);
  elu_inplace_kernel<<<nblk((long long)N * HC1), 256, 0, stream>>>(
      out1, (long long)N * HC1);

  // ---------------- layer 2 ----------------
  pad_b_kernel<<<nblk((long long)HC1 * 16), 256, 0, stream>>>(W2, W2p, HC1, NC,
                                                              16);
  {
    dim3 grid((tilesM + 7) / 8, 1);  // NB=1 -> one padded 16-col strip
    gemm_wmma_lds<1, 256><<<grid, 256, 0, stream>>>(out1, W2p, h2, N, 16, NC,
                                                    NC);
  }
  att_dots_kernel<<<nblk((long long)N), 256, 0, stream>>>(h2, as2, ad2, aS2,
                                                          aD2, N, 1, NC);
  fill_i32_kernel<<<nblk((long long)N), 256, 0, stream>>>(m2, mInit, (long long)N);
  fill_f32_kernel<<<nblk((long long)N), 256, 0, stream>>>(s2, 0.f, (long long)N);
  init_bias_kernel<<<nblk((long long)N * NC), 256, 0, stream>>>(
      out2, b2, NC, (long long)N * NC);

  edge_logit_max_kernel<<<nblk((long long)Etot), 256, 0, stream>>>(
      ei, E, N, aS2, aD2, m2, 1);
  edge_exp_sum_kernel<<<nblk((long long)Etot), 256, 0, stream>>>(
      ei, E, N, aS2, aD2, m2, e2, s2, 1);
  edge_aggregate_kernel<<<nblk((long long)Etot * NC), 256, 0, stream>>>(
      ei, E, N, e2, s2, h2, out2, 1, NC);

  (void)n_in; (void)out_size; (void)ws_size;
}